// GridTorch_81862076661764
// MI455X (gfx1250) — compile-verified
//
#include <hip/hip_runtime.h>
#include <hip/hip_bf16.h>

typedef __attribute__((ext_vector_type(2))) float v2f;
typedef __attribute__((ext_vector_type(8))) float v8f;

#define T_STEPS 100
#define BATCH   2048
#define NI      3
#define NU      128
#define NB      256
#define NPC     256
#define NHD     12

#define BT  64      // batch rows per recurrence block
#define HS  132     // LDS row stride (floats) for h / rh / states tiles: 132 % 64 == 4 -> conflict-free A-frags
#define US  136     // LDS row stride (floats) for U matrices: 2*136 % 64 == 16 -> conflict-free B-frags
#define BS  260     // LDS row stride for bottleneck tile in heads kernel (260 % 64 == 4)

__device__ __forceinline__ v8f zero8() {
    v8f z;
#pragma unroll
    for (int i = 0; i < 8; ++i) z[i] = 0.f;
    return z;
}

__device__ __forceinline__ v8f wmma_f32(v2f a, v2f b, v8f c) {
    // D = A(16x4,f32) * B(4x16,f32) + C(16x16,f32)
    return __builtin_amdgcn_wmma_f32_16x16x4_f32(false, a, false, b, (short)0, c, false, false);
}

// Fast transcendental epilogue: v_exp_f32 + v_rcp_f32 only (no IEEE div sequence).
__device__ __forceinline__ float sigmoid_fast(float x) {
    return __builtin_amdgcn_rcpf(1.f + __expf(-x));
}
__device__ __forceinline__ float tanh_fast(float x) {
    // tanh(x) = 1 - 2/(exp(2x)+1)
    return 1.f - 2.f * __builtin_amdgcn_rcpf(1.f + __expf(2.f * x));
}

// ---------------------------------------------------------------------------
// h0 = concat(init_pc, init_hd) @ W_embed + b_embed        [B, NU]
// ---------------------------------------------------------------------------
__global__ __launch_bounds__(256) void embed_kernel(
    const float* __restrict__ init_pc, const float* __restrict__ init_hd,
    const float* __restrict__ We, const float* __restrict__ be,
    float* __restrict__ h0) {
    const int idx = blockIdx.x * 256 + threadIdx.x;   // B*NU threads
    const int b = idx >> 7;
    const int j = idx & 127;
    float acc = be[j];
    const float* pc = init_pc + (size_t)b * NPC;
    const float* hd = init_hd + (size_t)b * NHD;
    for (int k = 0; k < NPC; ++k)  acc += pc[k] * We[k * NU + j];
    for (int k = 0; k < NHD; ++k)  acc += hd[k] * We[(NPC + k) * NU + j];
    h0[idx] = acc;
}

// ---------------------------------------------------------------------------
// GRU recurrence: each block owns a BT-row batch tile for all T steps.
// All 3 U matrices + h + r*h tiles live in the 320KB WGP LDS.
// Wave w (of 8) owns hidden columns [16w, 16w+16).
// ---------------------------------------------------------------------------
__global__ __launch_bounds__(256) void gru_recurrence(
    const float* __restrict__ x,                       // [T, B, NI]
    const float* __restrict__ h0,                      // [B, NU]
    const float* __restrict__ Wz, const float* __restrict__ Wr, const float* __restrict__ Wh,  // [NI, NU]
    const float* __restrict__ Uz, const float* __restrict__ Ur, const float* __restrict__ Uh,  // [NU, NU]
    const float* __restrict__ bz, const float* __restrict__ br, const float* __restrict__ bh,  // [NU]
    float* __restrict__ states,                        // [T, B, NU]  (d_out region)
    float* __restrict__ finalstate)                    // [B, NU]     (d_out region)
{
    extern __shared__ float smem[];
    float* sUz = smem;                      // NU * US
    float* sUr = sUz + NU * US;
    float* sUh = sUr + NU * US;
    float* sH  = sUh + NU * US;             // BT * HS
    float* sRH = sH  + BT * HS;             // BT * HS
    float* sX  = sRH + BT * HS;             // 2 * BT * NI (double buffered)

    const int tid  = threadIdx.x;
    const int lane = tid & 31;
    const int wv   = tid >> 5;              // wave id 0..7 = owned column slice
    const int hi   = lane >> 4;             // lane half
    const int ln   = lane & 15;
    const int jc   = (wv << 4) + ln;        // owned hidden column 0..127
    const int row0 = blockIdx.x * BT;       // first batch row of this tile

    // Stage U matrices into LDS (padded row stride)
    for (int i = tid; i < NU * NU; i += 256) {
        const int r = i >> 7, c = i & 127;
        sUz[r * US + c] = Uz[i];
        sUr[r * US + c] = Ur[i];
        sUh[r * US + c] = Uh[i];
    }
    // Stage h0 tile
    for (int i = tid; i < BT * NU; i += 256) {
        const int r = i >> 7, c = i & 127;
        sH[r * HS + c] = h0[(size_t)(row0 + r) * NU + c];
    }
    // Stage x tile for t = 0
    for (int i = tid; i < BT * NI; i += 256)
        sX[i] = x[(size_t)row0 * NI + i];
    __syncthreads();

    // Per-lane input-projection weights (K = NI = 3 folded into the epilogue)
    const float wz0 = Wz[jc], wz1 = Wz[NU + jc], wz2 = Wz[2 * NU + jc], bzc = bz[jc];
    const float wr0 = Wr[jc], wr1 = Wr[NU + jc], wr2 = Wr[2 * NU + jc], brc = br[jc];
    const float wh0 = Wh[jc], wh1 = Wh[NU + jc], wh2 = Wh[2 * NU + jc], bhc = bh[jc];

    for (int t = 0; t < T_STEPS; ++t) {
        const float* xc = sX + (t & 1) * (BT * NI);

        // ---- phase A: z and r gates (shared A fragments) ----
        v8f accz[4], accr[4];
#pragma unroll
        for (int m = 0; m < 4; ++m) { accz[m] = zero8(); accr[m] = zero8(); }

#pragma unroll
        for (int k4 = 0; k4 < 32; ++k4) {
            const int k0 = (k4 << 2) + (hi << 1);
            v2f bzf, brf;
            bzf[0] = sUz[k0 * US + jc]; bzf[1] = sUz[(k0 + 1) * US + jc];
            brf[0] = sUr[k0 * US + jc]; brf[1] = sUr[(k0 + 1) * US + jc];
#pragma unroll
            for (int m = 0; m < 4; ++m) {
                const v2f a = *(const v2f*)&sH[(16 * m + ln) * HS + k0];
                accz[m] = wmma_f32(a, bzf, accz[m]);
                accr[m] = wmma_f32(a, brf, accr[m]);
            }
        }

        float zreg[4][8];
#pragma unroll
        for (int m = 0; m < 4; ++m) {
#pragma unroll
            for (int v = 0; v < 8; ++v) {
                const int lrow = 16 * m + (hi << 3) + v;
                const float x0 = xc[lrow * NI + 0];
                const float x1 = xc[lrow * NI + 1];
                const float x2 = xc[lrow * NI + 2];
                const float zi = accz[m][v] + x0 * wz0 + x1 * wz1 + x2 * wz2 + bzc;
                const float ri = accr[m][v] + x0 * wr0 + x1 * wr1 + x2 * wr2 + brc;
                const float z  = sigmoid_fast(zi);
                const float r  = sigmoid_fast(ri);
                const float h  = sH[lrow * HS + jc];
                sRH[lrow * HS + jc] = r * h;
                zreg[m][v] = z;
            }
        }
        __syncthreads();   // r*h tile complete

        // Stage x tile for t+1 into the other buffer (consumed after next barrier)
        if (t + 1 < T_STEPS) {
            const float* xn = x + ((size_t)(t + 1) * BATCH + row0) * NI;
            float* xb = sX + ((t + 1) & 1) * (BT * NI);
            for (int i = tid; i < BT * NI; i += 256) xb[i] = xn[i];
        }

        // ---- phase B: candidate h~ = tanh(xh + (r*h) @ Uh) ----
        v8f acch[4];
#pragma unroll
        for (int m = 0; m < 4; ++m) acch[m] = zero8();

#pragma unroll
        for (int k4 = 0; k4 < 32; ++k4) {
            const int k0 = (k4 << 2) + (hi << 1);
            v2f bhf;
            bhf[0] = sUh[k0 * US + jc]; bhf[1] = sUh[(k0 + 1) * US + jc];
#pragma unroll
            for (int m = 0; m < 4; ++m) {
                const v2f a = *(const v2f*)&sRH[(16 * m + ln) * HS + k0];
                acch[m] = wmma_f32(a, bhf, acch[m]);
            }
        }

#pragma unroll
        for (int m = 0; m < 4; ++m) {
#pragma unroll
            for (int v = 0; v < 8; ++v) {
                const int lrow = 16 * m + (hi << 3) + v;
                const float x0 = xc[lrow * NI + 0];
                const float x1 = xc[lrow * NI + 1];
                const float x2 = xc[lrow * NI + 2];
                const float hti = acch[m][v] + x0 * wh0 + x1 * wh1 + x2 * wh2 + bhc;
                const float ht  = tanh_fast(hti);
                const float z   = zreg[m][v];
                const float hold = sH[lrow * HS + jc];   // (lrow,jc) only touched by this lane
                const float hn  = (1.f - z) * hold + z * ht;
                sH[lrow * HS + jc] = hn;   // own columns only: no cross-wave hazard
                states[((size_t)t * BATCH + row0 + lrow) * NU + jc] = hn;
            }
        }
        __syncthreads();   // h fully updated before next step reads all columns
    }

    // final_state = h after last step
    for (int i = tid; i < BT * NU; i += 256) {
        const int r = i >> 7, c = i & 127;
        finalstate[(size_t)(row0 + r) * NU + c] = sH[r * HS + c];
    }
}

// ---------------------------------------------------------------------------
// Fused heads over T*B rows: bneck = states @ W_bneck;
// logits_pc = bneck @ W_pc + b_pc; logits_hd = bneck @ W_hd + b_hd.
// 64-row tiles; wave w owns output column slices {w, w+8}.
// ---------------------------------------------------------------------------
__global__ __launch_bounds__(256) void heads_kernel(
    const float* __restrict__ states,                  // [T*B, NU]
    const float* __restrict__ Wb,                      // [NU, NB]
    const float* __restrict__ Wpc, const float* __restrict__ bpc,  // [NB, NPC], [NPC]
    const float* __restrict__ Whd, const float* __restrict__ bhd,  // [NB, NHD], [NHD]
    float* __restrict__ out_bn, float* __restrict__ out_pc, float* __restrict__ out_hd)
{
    extern __shared__ float smem[];
    float* sS = smem;               // 64 * HS  (states tile)
    float* sB = sS + 64 * HS;       // 64 * BS  (bottleneck tile)

    const int tid  = threadIdx.x;
    const int lane = tid & 31;
    const int wv   = tid >> 5;
    const int hi   = lane >> 4;
    const int ln   = lane & 15;
    const size_t r0 = (size_t)blockIdx.x * 64;

    for (int i = tid; i < 64 * NU; i += 256) {
        const int r = i >> 7, c = i & 127;
        sS[r * HS + c] = states[(r0 + r) * NU + c];
    }
    __syncthreads();

    const int j0 = (wv << 4) + ln;          // first owned NB column
    const int j1 = ((wv + 8) << 4) + ln;    // second owned NB column

    // ---- bottleneck: K = 128 ----
    v8f acc0[4], acc1[4];
#pragma unroll
    for (int m = 0; m < 4; ++m) { acc0[m] = zero8(); acc1[m] = zero8(); }

#pragma unroll
    for (int k4 = 0; k4 < 32; ++k4) {
        const int k0 = (k4 << 2) + (hi << 1);
        v2f b0, b1;
        b0[0] = Wb[k0 * NB + j0]; b0[1] = Wb[(k0 + 1) * NB + j0];
        b1[0] = Wb[k0 * NB + j1]; b1[1] = Wb[(k0 + 1) * NB + j1];
#pragma unroll
        for (int m = 0; m < 4; ++m) {
            const v2f a = *(const v2f*)&sS[(16 * m + ln) * HS + k0];
            acc0[m] = wmma_f32(a, b0, acc0[m]);
            acc1[m] = wmma_f32(a, b1, acc1[m]);
        }
    }

#pragma unroll
    for (int m = 0; m < 4; ++m) {
#pragma unroll
        for (int v = 0; v < 8; ++v) {
            const int lrow = 16 * m + (hi << 3) + v;
            const float v0 = acc0[m][v];
            const float v1 = acc1[m][v];
            out_bn[(r0 + lrow) * NB + j0] = v0;
            out_bn[(r0 + lrow) * NB + j1] = v1;
            sB[lrow * BS + j0] = v0;
            sB[lrow * BS + j1] = v1;
        }
    }
    __syncthreads();

    // ---- logits_pc: K = 256 ----
    v8f accp0[4], accp1[4];
#pragma unroll
    for (int m = 0; m < 4; ++m) { accp0[m] = zero8(); accp1[m] = zero8(); }

#pragma unroll
    for (int k4 = 0; k4 < 64; ++k4) {
        const int k0 = (k4 << 2) + (hi << 1);
        v2f b0, b1;
        b0[0] = Wpc[k0 * NPC + j0]; b0[1] = Wpc[(k0 + 1) * NPC + j0];
        b1[0] = Wpc[k0 * NPC + j1]; b1[1] = Wpc[(k0 + 1) * NPC + j1];
#pragma unroll
        for (int m = 0; m < 4; ++m) {
            const v2f a = *(const v2f*)&sB[(16 * m + ln) * BS + k0];
            accp0[m] = wmma_f32(a, b0, accp0[m]);
            accp1[m] = wmma_f32(a, b1, accp1[m]);
        }
    }

    const float bp0 = bpc[j0];
    const float bp1 = bpc[j1];
#pragma unroll
    for (int m = 0; m < 4; ++m) {
#pragma unroll
        for (int v = 0; v < 8; ++v) {
            const int lrow = 16 * m + (hi << 3) + v;
            out_pc[(r0 + lrow) * NPC + j0] = accp0[m][v] + bp0;
            out_pc[(r0 + lrow) * NPC + j1] = accp1[m][v] + bp1;
        }
    }

    // ---- logits_hd: N = 12 (single masked 16-wide tile); waves 0..3 take one m-tile each ----
    if (wv < 4) {
        const int m = wv;
        v8f acch = zero8();
#pragma unroll
        for (int k4 = 0; k4 < 64; ++k4) {
            const int k0 = (k4 << 2) + (hi << 1);
            v2f b;
            b[0] = (ln < NHD) ? Whd[k0 * NHD + ln] : 0.f;
            b[1] = (ln < NHD) ? Whd[(k0 + 1) * NHD + ln] : 0.f;
            const v2f a = *(const v2f*)&sB[(16 * m + ln) * BS + k0];
            acch = wmma_f32(a, b, acch);
        }
        if (ln < NHD) {
            const float bb = bhd[ln];
#pragma unroll
            for (int v = 0; v < 8; ++v) {
                const int lrow = 16 * m + (hi << 3) + v;
                out_hd[(r0 + lrow) * NHD + ln] = acch[v] + bb;
            }
        }
    }
}

// ---------------------------------------------------------------------------
extern "C" void kernel_launch(void* const* d_in, const int* in_sizes, int n_in,
                              void* d_out, int out_size, void* d_ws, size_t ws_size,
                              hipStream_t stream) {
    const float* x      = (const float*)d_in[0];
    const float* initpc = (const float*)d_in[1];
    const float* inithd = (const float*)d_in[2];
    const float* Wz     = (const float*)d_in[3];
    const float* Wr     = (const float*)d_in[4];
    const float* Wh     = (const float*)d_in[5];
    const float* Uz     = (const float*)d_in[6];
    const float* Ur     = (const float*)d_in[7];
    const float* Uh     = (const float*)d_in[8];
    const float* bz     = (const float*)d_in[9];
    const float* br     = (const float*)d_in[10];
    const float* bh     = (const float*)d_in[11];
    const float* We     = (const float*)d_in[12];
    const float* be     = (const float*)d_in[13];
    const float* Wb     = (const float*)d_in[14];
    const float* Wpc    = (const float*)d_in[15];
    const float* bpc    = (const float*)d_in[16];
    const float* Whd    = (const float*)d_in[17];
    const float* bhd    = (const float*)d_in[18];
    (void)in_sizes; (void)n_in; (void)out_size; (void)ws_size;

    float* out    = (float*)d_out;
    float* out_hd = out;
    float* out_pc = out_hd + (size_t)T_STEPS * BATCH * NHD;
    float* out_bn = out_pc + (size_t)T_STEPS * BATCH * NPC;
    float* out_st = out_bn + (size_t)T_STEPS * BATCH * NB;
    float* out_fs = out_st + (size_t)T_STEPS * BATCH * NU;

    float* h0 = (float*)d_ws;   // [B, NU] scratch

    embed_kernel<<<(BATCH * NU) / 256, 256, 0, stream>>>(initpc, inithd, We, be, h0);

    const size_t lds_rec = (size_t)(3 * NU * US + 2 * BT * HS + 2 * BT * NI) * sizeof(float);
    gru_recurrence<<<BATCH / BT, 256, lds_rec, stream>>>(
        x, h0, Wz, Wr, Wh, Uz, Ur, Uh, bz, br, bh, out_st, out_fs);

    const size_t lds_heads = (size_t)(64 * HS + 64 * BS) * sizeof(float);
    heads_kernel<<<(T_STEPS * BATCH) / 64, 256, lds_heads, stream>>>(
        out_st, Wb, Wpc, bpc, Whd, bhd, out_bn, out_pc, out_hd);
}